// DistributedEmbedding_63488206569530
// MI455X (gfx1250) — compile-verified
//
#include <hip/hip_runtime.h>

// Problem constants (match the reference).
#define BATCH     16384
#define SLOT_NUM  26
#define EMB_DIM   64
#define NUM_ROWS  (BATCH * SLOT_NUM)       // 425,984 output rows
#define NNZ       (NUM_ROWS * 4)           // 1,703,936 keys

typedef float f4 __attribute__((ext_vector_type(4)));

// ---------------------------------------------------------------------------
// Kernel B (placed first so the disasm snippet shows it): gather + segment
// mean. One output row per 16-lane group (2 rows per wave32); lane L owns
// columns [4L, 4L+4) as a float4, so each gathered table row is a single
// contiguous 256 B request.
// ---------------------------------------------------------------------------
__global__ __launch_bounds__(256)
void embed_mean_kernel(const float* __restrict__ table,
                       const int*   __restrict__ values,
                       const int*   __restrict__ row_ptr,
                       float*       __restrict__ out)
{
    const int gtid = blockIdx.x * blockDim.x + threadIdx.x;
    const int row  = gtid >> 4;        // output row handled by this 16-lane group
    const int lane = gtid & 15;        // column chunk within the row
    if (row >= NUM_ROWS) return;

    // Segment boundaries: two coalesced loads (row_ptr is L2-resident, 1.7 MB).
    const int start = row_ptr[row];
    const int end   = row_ptr[row + 1];
    const int cnt   = end - start;

    const float* colbase = table + (lane << 2);   // this lane's 16 B column slice

    // Prefetch pass: pull this segment's table rows toward the WGP
    // (emits global_prefetch_b8; breaks the values->table dependent chain).
    for (int k = start; k < end; ++k) {
        const int idx = values[k];
        __builtin_prefetch(colbase + (size_t)idx * EMB_DIM, 0, 3);
    }

    // Gather + accumulate, unrolled x2 so two independent index loads and two
    // independent 256 B gathers are in flight per iteration (MLP within the
    // wave on top of the prefetches).
    f4 sum0 = (f4)0.0f;
    f4 sum1 = (f4)0.0f;
    int k = start;
    for (; k + 1 < end; k += 2) {
        const int idx0 = values[k];
        const int idx1 = values[k + 1];
        const f4 v0 = *(const f4*)(colbase + (size_t)idx0 * EMB_DIM);
        const f4 v1 = *(const f4*)(colbase + (size_t)idx1 * EMB_DIM);
        sum0 += v0;
        sum1 += v1;
    }
    if (k < end) {
        const int idx = values[k];
        sum0 += *(const f4*)(colbase + (size_t)idx * EMB_DIM);
    }
    f4 sum = sum0 + sum1;

    // Mean combiner; empty rows stay zero (denominator max(count,1)).
    const float scale = 1.0f / (float)(cnt > 0 ? cnt : 1);
    sum *= scale;

    // Streamed output, written once: non-temporal 128-bit store keeps the
    // 109 MB output from evicting reusable table lines in L2.
    f4* dst = (f4*)(out + (size_t)row * EMB_DIM + (lane << 2));
    __builtin_nontemporal_store(sum, dst);
}

// ---------------------------------------------------------------------------
// Kernel A: build CSR row pointers from the *sorted* row_indices (COO->CSR).
// Thread k writes row_ptr[row] = k for every row in (row_indices[k-1],
// row_indices[k]]; the virtual thread k==NNZ fills the tail with NNZ.
// Every entry of row_ptr[0..NUM_ROWS] is written exactly once -> no atomics,
// deterministic, and overwrites the 0xAA workspace poison completely.
// row_ptr[row] == lower_bound(row_indices, row); end(row) == row_ptr[row+1].
// ---------------------------------------------------------------------------
__global__ __launch_bounds__(256)
void build_row_ptr_kernel(const int* __restrict__ row_indices,
                          int*       __restrict__ row_ptr)
{
    const int k = blockIdx.x * blockDim.x + threadIdx.x;
    if (k > NNZ) return;

    const int prev = (k == 0)   ? -1       : row_indices[k - 1];
    const int cur  = (k == NNZ) ? NUM_ROWS : row_indices[k];

    // Fill the gap (prev, cur]; average trip count ~= 1 (sorted random rows).
    for (int row = prev + 1; row <= cur; ++row) {
        row_ptr[row] = k;
    }
}

extern "C" void kernel_launch(void* const* d_in, const int* in_sizes, int n_in,
                              void* d_out, int out_size, void* d_ws, size_t ws_size,
                              hipStream_t stream)
{
    (void)in_sizes; (void)n_in; (void)ws_size; (void)out_size;

    const float* table       = (const float*)d_in[0];
    const int*   values      = (const int*)d_in[1];
    const int*   row_indices = (const int*)d_in[2];
    float*       out         = (float*)d_out;
    int*         row_ptr     = (int*)d_ws;      // (NUM_ROWS+1) ints ~= 1.7 MB

    // Pass A: COO -> CSR row pointers (NNZ+1 logical threads).
    {
        const int n     = NNZ + 1;
        const int block = 256;
        const int grid  = (n + block - 1) / block;
        build_row_ptr_kernel<<<grid, block, 0, stream>>>(row_indices, row_ptr);
    }

    // Pass B: gather + segment mean (16 lanes per output row).
    {
        const int total_threads = NUM_ROWS * 16;
        const int block = 256;                   // 8 wave32s
        const int grid  = (total_threads + block - 1) / block;
        embed_mean_kernel<<<grid, block, 0, stream>>>(table, values, row_ptr, out);
    }
}